// SpatialAttention_18674517803461
// MI455X (gfx1250) — compile-verified
//
#include <hip/hip_runtime.h>
#include <hip/hip_bf16.h>

// Spatial attention, fused, f32 WMMA (V_WMMA_F32_16X16X4_F32) +
// double-buffered GLOBAL_LOAD_ASYNC_TO_LDS_B128 staging of the x stream.
// Shapes: x[16,64,64,512], sentence[16,256] (UNUSED by reference),
// context[16,64,256], mask[16,64] (int, nonzero => masked -> -inf),
// W_word[256,512]. Outputs concatenated: weighted[16,4096,512] then attn[16,4096,64].

typedef float v2f __attribute__((ext_vector_type(2)));
typedef float v4f __attribute__((ext_vector_type(4)));
typedef float v8f __attribute__((ext_vector_type(8)));

#define BSZ   16
#define HW    4096
#define CDIM  512
#define SEQ   64
#define EMB   256

#define CTX_PITCH 516    // pitch%64==4 -> conflict-free column reads; rows 16B aligned
#define P_PITCH   68     // pitch%64==4; rows 16B aligned
#define X_PITCH   68     // staging pitch: banks (4*row+k) distinct, 16B rows for B128
#define XCHUNK_F  (16 * X_PITCH)   // one staged chunk: 16 rows x 64 cols (padded)

static __device__ __forceinline__ v8f wmma_f32(v2f a, v2f b, v8f c) {
  // D = A(16x4) * B(4x16) + C(16x16), all f32.
  return __builtin_amdgcn_wmma_f32_16x16x4_f32(false, a, false, b, (short)0, c,
                                               false, false);
}

// Async DMA: 16B per lane, global -> LDS, tracked on ASYNCcnt.
static __device__ __forceinline__ void async_copy_b128(unsigned lds_byte_off,
                                                       const float* gaddr) {
  asm volatile("global_load_async_to_lds_b128 %0, %1, off"
               :: "v"(lds_byte_off), "v"(gaddr)
               : "memory");
}

__launch_bounds__(256, 1)
__global__ void spatial_attention_fused(const float* __restrict__ x,
                                        const float* __restrict__ context,
                                        const int*   __restrict__ mask,
                                        const float* __restrict__ Wword,
                                        float* __restrict__ outW,
                                        float* __restrict__ outA) {
  __shared__ float ctxL[SEQ * CTX_PITCH];        // projected word features, 64x512
  __shared__ float Pbuf[8 * 16 * P_PITCH];       // per-wave score/prob tiles
  __shared__ float Xbuf[8 * 2 * XCHUNK_F];       // per-wave double-buffered x staging
  __shared__ float maskAdd[SEQ];                 // 0 or -inf per word

  const int tid    = threadIdx.x;
  const int wave   = tid >> 5;
  const int lane   = tid & 31;
  const int lane16 = lane & 15;
  const int h      = lane >> 4;                  // half-wave: K pair select
  const int b      = blockIdx.x;
  const int posBase = blockIdx.y * 256;          // 16 blocks x 256 positions

  if (tid < SEQ)
    maskAdd[tid] = mask[b * SEQ + tid] ? -INFINITY : 0.0f;

  // ---------- Phase 1: ctx = context[b] @ W_word into LDS ----------
  {
    const int m0    = (wave >> 1) * 16;          // seq row tile
    const int nBase = (wave & 1) * 256;          // channel half
    const float* cg = context + (size_t)b * SEQ * EMB;
    for (int chunk = 0; chunk < 4; ++chunk) {
      const int n0 = nBase + chunk * 64;
      v8f acc[4] = {};
      for (int ks = 0; ks < EMB / 4; ++ks) {
        const int k0 = ks * 4;
        v2f a = *(const v2f*)(cg + (m0 + lane16) * EMB + k0 + 2 * h);
#pragma unroll
        for (int t = 0; t < 4; ++t) {
          const float* bp = Wword + (size_t)(k0 + 2 * h) * CDIM + n0 + t * 16 + lane16;
          v2f bf = {bp[0], bp[CDIM]};
          acc[t] = wmma_f32(a, bf, acc[t]);
        }
      }
#pragma unroll
      for (int t = 0; t < 4; ++t)
#pragma unroll
        for (int r = 0; r < 8; ++r)
          ctxL[(m0 + h * 8 + r) * CTX_PITCH + n0 + t * 16 + lane16] = acc[t][r];
    }
  }
  __syncthreads();

  float* P = Pbuf + wave * (16 * P_PITCH);
  float* xstage = Xbuf + wave * (2 * XCHUNK_F);
  const unsigned xs_lds0 = (unsigned)(size_t)(void*)xstage;  // LDS byte address

  for (int it = 0; it < 2; ++it) {
    const int row0 = posBase + it * 128 + wave * 16;   // this wave's 16 positions
    const float* xrow = x + ((size_t)b * HW + row0) * CDIM;

    if (it == 0)  // warm L2 for next iteration tile (global_prefetch_b8)
      __builtin_prefetch(xrow + (size_t)128 * CDIM + lane * 16, 0, 0);

    // Stage one 16x64 chunk of x: 256 b128 units, 8 per lane (rows 2i, 2i+1).
#define ISSUE_CHUNK(KC, BUFSEL)                                                 \
    {                                                                           \
      _Pragma("unroll")                                                         \
      for (int i = 0; i < 8; ++i) {                                             \
        const int u  = lane + 32 * i;                                           \
        const int r  = u >> 4;                                                  \
        const int c4 = (u & 15) * 4;                                            \
        const float* g = xrow + (size_t)r * CDIM + (KC) * 64 + c4;              \
        const unsigned l = xs_lds0 +                                            \
            (unsigned)(((BUFSEL) * XCHUNK_F + r * X_PITCH + c4) * 4);           \
        async_copy_b128(l, g);                                                  \
      }                                                                         \
    }

    // ---------- Phase 2: scores S = X(16x512) @ ctx^T(512x64) ----------
    asm volatile("s_wait_dscnt 0x0" ::: "memory");   // staging buffers reusable
    ISSUE_CHUNK(0, 0);
    v8f sacc[4] = {};
    for (int kc = 0; kc < 8; ++kc) {
      if (kc < 7) {
        ISSUE_CHUNK(kc + 1, (kc + 1) & 1);
        asm volatile("s_wait_asynccnt 0x8" ::: "memory");  // chunk kc landed
      } else {
        asm volatile("s_wait_asynccnt 0x0" ::: "memory");
      }
      const float* xs = xstage + (kc & 1) * XCHUNK_F;
#pragma unroll
      for (int ks = 0; ks < 16; ++ks) {
        const int k0 = kc * 64 + ks * 4;
        v2f a = *(const v2f*)(xs + lane16 * X_PITCH + ks * 4 + 2 * h);
#pragma unroll
        for (int t = 0; t < 4; ++t) {
          // B[k][n=s] = ctx[s][k]; padded pitch keeps the 16 rows on distinct banks
          v2f bf = *(const v2f*)(&ctxL[(t * 16 + lane16) * CTX_PITCH + k0 + 2 * h]);
          sacc[t] = wmma_f32(a, bf, sacc[t]);
        }
      }
    }
#undef ISSUE_CHUNK

    // spill score tiles to per-wave LDS (D tile: vgpr r -> row r / 8+r)
#pragma unroll
    for (int t = 0; t < 4; ++t)
#pragma unroll
      for (int r = 0; r < 8; ++r)
        P[(h * 8 + r) * P_PITCH + t * 16 + lane16] = sacc[t][r];

    // ---------- Phase 3: masked softmax over SEQ=64 ----------
    // lane handles row lane16, columns [32h, 32h+32); halves combine via shfl_xor(16)
    float vals[32];
    float* prow = P + lane16 * P_PITCH + h * 32;
    float mx = -INFINITY;
#pragma unroll
    for (int j = 0; j < 32; ++j) {
      float v = prow[j] + maskAdd[h * 32 + j];   // masked -> -inf
      vals[j] = v;
      mx = fmaxf(mx, v);
    }
    mx = fmaxf(mx, __shfl_xor(mx, 16, 32));
    float sum = 0.0f;
#pragma unroll
    for (int j = 0; j < 32; ++j) { vals[j] = __expf(vals[j] - mx); sum += vals[j]; }
    sum += __shfl_xor(sum, 16, 32);
    const float inv = 1.0f / sum;
#pragma unroll
    for (int j = 0; j < 32; ++j) vals[j] *= inv;
#pragma unroll
    for (int j = 0; j < 32; ++j) prow[j] = vals[j];   // probs back to LDS for A-frags
    {
      float* arow = outA + ((size_t)b * HW + row0 + lane16) * SEQ + h * 32;
#pragma unroll
      for (int j = 0; j < 32; j += 4) {
        v4f vv = {vals[j], vals[j + 1], vals[j + 2], vals[j + 3]};
        *(v4f*)(arow + j) = vv;
      }
    }

    // ---------- Phase 4: weighted = P(16x64) @ ctx(64x512) ----------
    v2f afrag[16];                                    // all A-frags cached (K=64)
#pragma unroll
    for (int ks = 0; ks < 16; ++ks)
      afrag[ks] = *(const v2f*)(P + lane16 * P_PITCH + ks * 4 + 2 * h);

    for (int nc = 0; nc < 8; ++nc) {
      const int n0 = nc * 64;
      v8f wacc[4] = {};
#pragma unroll
      for (int ks = 0; ks < 16; ++ks) {
        const int kr = ks * 4 + 2 * h;
#pragma unroll
        for (int t = 0; t < 4; ++t) {
          const int nn = n0 + t * 16 + lane16;
          v2f bf = {ctxL[kr * CTX_PITCH + nn], ctxL[(kr + 1) * CTX_PITCH + nn]};
          wacc[t] = wmma_f32(afrag[ks], bf, wacc[t]);
        }
      }
#pragma unroll
      for (int t = 0; t < 4; ++t)
#pragma unroll
        for (int r = 0; r < 8; ++r)
          outW[((size_t)b * HW + row0 + h * 8 + r) * CDIM + n0 + t * 16 + lane16] =
              wacc[t][r];
    }
  }
}

extern "C" void kernel_launch(void* const* d_in, const int* in_sizes, int n_in,
                              void* d_out, int out_size, void* d_ws, size_t ws_size,
                              hipStream_t stream) {
  (void)in_sizes; (void)n_in; (void)out_size; (void)d_ws; (void)ws_size;
  const float* x       = (const float*)d_in[0];
  // d_in[1] = sentence: unused by the reference computation
  const float* context = (const float*)d_in[2];
  const int*   mask    = (const int*)d_in[3];   // bool -> int per harness convention
  const float* Wword   = (const float*)d_in[4];
  float* outW = (float*)d_out;                               // [16,4096,512]
  float* outA = outW + (size_t)BSZ * HW * CDIM;              // [16,4096,64]

  dim3 grid(BSZ, 16);   // batch x position-block (256 positions each)
  dim3 block(256);      // 8 wave32
  hipLaunchKernelGGL(spatial_attention_fused, grid, block, 0, stream,
                     x, context, mask, Wword, outW, outA);
}